// PaletteRenderer_39393440039527
// MI455X (gfx1250) — compile-verified
//
#include <hip/hip_runtime.h>
#include <hip/hip_bf16.h>
#include <math.h>

typedef float v2f __attribute__((ext_vector_type(2)));
typedef float v8f __attribute__((ext_vector_type(8)));

#define NB 6
#define CLIPD 16
#define NCH 40           // image(3) depth(1) wsum(1) all_map(35)
#define NCHP 48          // padded to 3 x 16 WMMA column tiles
#define BLK 256

// ---------------------------------------------------------------------------
// Pass 1: per-ray sample ranges via lower_bound on sorted segment ids.
// starts[r] = first i with seg[i] >= r ; starts[N] = M.
// ---------------------------------------------------------------------------
__global__ __launch_bounds__(256) void starts_kernel(
    const int* __restrict__ seg, int M, int N, int* __restrict__ starts) {
  int r = blockIdx.x * blockDim.x + threadIdx.x;
  if (r > N) return;
  if (r == N) { starts[N] = M; return; }
  int lo = 0, hi = M;
  while (lo < hi) {
    int mid = (lo + hi) >> 1;
    if (seg[mid] < r) lo = mid + 1; else hi = mid;
  }
  starts[r] = lo;
}

__device__ __forceinline__ float softplusf(float x) {
  return (x > 20.f) ? x : log1pf(expf(x));
}

// ---------------------------------------------------------------------------
// Pass 2: one workgroup (8 wave32) per ray.
// ---------------------------------------------------------------------------
__global__ __launch_bounds__(BLK) void palette_main(
    const float* __restrict__ sigmas,   const float* __restrict__ radiance,
    const float* __restrict__ offsets,  const float* __restrict__ omega,
    const float* __restrict__ view_dep, const float* __restrict__ diffuse,
    const float* __restrict__ clip_feat,const float* __restrict__ basis_color,
    const float* __restrict__ dt,       const float* __restrict__ ts,
    const float* __restrict__ nears,    const float* __restrict__ fars,
    const int* __restrict__ starts,     float* __restrict__ out) {
  const int ray = blockIdx.x;
  const int tid = threadIdx.x;

  __shared__ float s_scan[BLK];
  __shared__ float s_pw[8 * NCHP];   // per-wave partial sums, padded channels
  __shared__ float s_final[NCHP];

  for (int j = tid; j < 8 * NCHP; j += BLK) s_pw[j] = 0.f;

  // clamp(basis_color, 0, 1) : 18 floats, L2-resident, load to registers once
  float bcc[NB * 3];
#pragma unroll
  for (int j = 0; j < NB * 3; ++j) {
    float v = basis_color[j];
    bcc[j] = fminf(fmaxf(v, 0.f), 1.f);
  }

  const int s0 = starts[ray];
  const int s1 = starts[ray + 1];
  const int nchunks = (s1 - s0 + BLK - 1) >> 8;

  float acc[NCH];
#pragma unroll
  for (int c = 0; c < NCH; ++c) acc[c] = 0.f;

  float carry = 0.f;   // running optical depth of previous chunks (this ray)
  __syncthreads();

  for (int ch = 0; ch < nchunks; ++ch) {
    const int i = s0 + (ch << 8) + tid;
    const bool valid = i < s1;

    // prefetch next chunk's two largest streams (global_prefetch_b8)
    if (valid && (i + BLK) < s1) {
      __builtin_prefetch(offsets  + (size_t)(i + BLK) * (NB * 3), 0, 0);
      __builtin_prefetch(clip_feat + (size_t)(i + BLK) * CLIPD, 0, 0);
    }

    float neglog = 0.f;
    if (valid) neglog = sigmas[i] * dt[i];

    // block-wide inclusive scan (Hillis-Steele) of neglog
    s_scan[tid] = neglog;
    __syncthreads();
#pragma unroll
    for (int st = 1; st < BLK; st <<= 1) {
      float v = (tid >= st) ? s_scan[tid - st] : 0.f;
      __syncthreads();
      s_scan[tid] += v;
      __syncthreads();
    }
    const float incl  = s_scan[tid];
    const float total = s_scan[BLK - 1];
    __syncthreads();  // protect s_scan before next chunk overwrites

    const float excl = carry + incl - neglog;  // exclusive per-ray optical depth
    carry += total;

    float w = 0.f;
    if (valid) {
      float T = expf(-excl);
      if (T > 1e-4f) w = (1.f - expf(-neglog)) * T;
    }

    if (w > 0.f) {
      const size_t ii = (size_t)i;
      const float sp = softplusf(radiance[ii]);

      float om[NB], om_s = 0.f, om_s2 = 0.f;
#pragma unroll
      for (int b = 0; b < NB; ++b) {
        float o = omega[ii * NB + b];
        om[b] = o; om_s += o; om_s2 += o * o;
      }

      float off[NB * 3], off_n = 0.f;
#pragma unroll
      for (int j = 0; j < NB * 3; ++j) {
        float o = offsets[ii * (NB * 3) + j];
        off[j] = o; off_n += o * o;
      }

      float vd[3], dif[3], vd_n = 0.f;
#pragma unroll
      for (int k = 0; k < 3; ++k) {
        float v = view_dep[ii * 3 + k];
        vd[k] = v; vd_n += v * v;
        dif[k] = diffuse[ii * 3 + k];
      }

      float rgb[3];
#pragma unroll
      for (int k = 0; k < 3; ++k) {
        float s = 0.f;
#pragma unroll
        for (int b = 0; b < NB; ++b) s += om[b] * (bcc[b * 3 + k] + off[b * 3 + k]);
        rgb[k] = sp * s + vd[k];
      }

      const float omega_sp = om_s / (om_s2 + 1e-6f) - 1.f;

      acc[0]  += w * rgb[0]; acc[1] += w * rgb[1]; acc[2] += w * rgb[2];
      acc[3]  += w * ts[ii];
      acc[4]  += w;
      acc[5]  += w * omega_sp;
      acc[6]  += w * vd_n;
      acc[7]  += w * off_n;
      // acc[8] : smooth_norm == 0
      acc[9]  += w * vd[0];  acc[10] += w * vd[1];  acc[11] += w * vd[2];
      acc[12] += w * (dif[0] + vd[0]);
      acc[13] += w * (dif[1] + vd[1]);
      acc[14] += w * (dif[2] + vd[2]);
      acc[15] += w * dif[0]; acc[16] += w * dif[1]; acc[17] += w * dif[2];

      const float4* cf = reinterpret_cast<const float4*>(clip_feat + ii * CLIPD);
#pragma unroll
      for (int q = 0; q < 4; ++q) {
        float4 c4 = cf[q];
        acc[18 + 4 * q + 0] += w * c4.x;
        acc[18 + 4 * q + 1] += w * c4.y;
        acc[18 + 4 * q + 2] += w * c4.z;
        acc[18 + 4 * q + 3] += w * c4.w;
      }
#pragma unroll
      for (int b = 0; b < NB; ++b) acc[34 + b] += w * om[b];
    }
  }

  // ---- wave32 butterfly reduction of 40 channels; lane 0 -> LDS partials ----
  const int wave = tid >> 5;
  const int lane = tid & 31;
#pragma unroll
  for (int c = 0; c < NCH; ++c) {
    float v = acc[c];
    v += __shfl_xor(v, 16, 32);
    v += __shfl_xor(v, 8, 32);
    v += __shfl_xor(v, 4, 32);
    v += __shfl_xor(v, 2, 32);
    v += __shfl_xor(v, 1, 32);
    if (lane == 0) s_pw[wave * NCHP + c] = v;
  }
  __syncthreads();

  // ---- wave 0: colsum of P(8 x 48) via V_WMMA_F32_16X16X4_F32 ----
  // D = ones(16x4) * B(4x16) + C, accumulated over two 4-row groups.
  // A = all ones is layout-invariant; any K-slot permutation of B rows still
  // yields exact per-column sums (full f32 precision).
  if (tid < 32) {
    const int col  = lane & 15;
    const int half = lane >> 4;
    v2f a; a.x = 1.0f; a.y = 1.0f;
#pragma unroll
    for (int t = 0; t < 3; ++t) {
      const int c0 = t * 16;
      v8f c8 = {};
#pragma unroll
      for (int g = 0; g < 2; ++g) {
        v2f b;
        b.x = s_pw[(4 * g + 2 * half + 0) * NCHP + c0 + col];
        b.y = s_pw[(4 * g + 2 * half + 1) * NCHP + c0 + col];
        c8 = __builtin_amdgcn_wmma_f32_16x16x4_f32(
            false, a, false, b, (short)0, c8, false, false);
      }
      if (half == 0) s_final[c0 + col] = c8[0];  // row M=0 (all rows identical)
    }

    // ---- background blend + depth normalization, write 40 outputs ----
    const float wsum  = s_final[4];
    const float bgadd = 1.0f - wsum;   // bg == 1.0
    const float nr = nears[ray];
    const float fr = fars[ray];
    for (int c = lane; c < NCH; c += 32) {
      float v = s_final[c];
      if (c < 3)                 v += bgadd;                       // image
      else if (c == 3)           v = fmaxf(v - nr, 0.f) / (fr - nr); // depth
      else if (c >= 12 && c < 15) v += bgadd;                      // direct_rgb_map
      out[(size_t)ray * NCH + c] = v;
    }
  }
}

// ---------------------------------------------------------------------------
extern "C" void kernel_launch(void* const* d_in, const int* in_sizes, int n_in,
                              void* d_out, int out_size, void* d_ws, size_t ws_size,
                              hipStream_t stream) {
  const float* sigmas      = (const float*)d_in[0];
  const float* radiance    = (const float*)d_in[1];
  const float* offsets     = (const float*)d_in[2];
  const float* omega       = (const float*)d_in[3];
  const float* view_dep    = (const float*)d_in[4];
  const float* diffuse     = (const float*)d_in[5];
  const float* clip_feat   = (const float*)d_in[6];
  const float* basis_color = (const float*)d_in[7];
  const float* dt          = (const float*)d_in[8];
  const float* ts          = (const float*)d_in[9];
  const float* nears       = (const float*)d_in[10];
  const float* fars        = (const float*)d_in[11];
  const int*   seg         = (const int*)d_in[12];

  const int M = in_sizes[0];
  const int N = in_sizes[10];

  int*   starts = (int*)d_ws;      // (N+1) ints
  float* out    = (float*)d_out;   // N x 40 floats

  starts_kernel<<<dim3((N + 1 + 255) / 256), dim3(256), 0, stream>>>(seg, M, N, starts);

  palette_main<<<dim3(N), dim3(BLK), 0, stream>>>(
      sigmas, radiance, offsets, omega, view_dep, diffuse, clip_feat,
      basis_color, dt, ts, nears, fars, starts, out);
}